// EMACodebook_45870250721585
// MI455X (gfx1250) — compile-verified
//
#include <hip/hip_runtime.h>

// ---------------------------------------------------------------------------
// VQ-VAE codebook lookup for MI455X (gfx1250), fp32 WMMA path.
// B=16, D=64, T=4096, K=1024.  BT = 65536 query rows.
// dist argmin computed as argmin_n ( e_sq[n] - 2 * <x, e_n> )  (x_sq constant/row)
// Outputs (concatenated float): z_q_st[4194304], commit[1], indices[65536],
//                               perplexity[1], usage_hist[1024]
// ---------------------------------------------------------------------------

#define BB   16
#define DD   64
#define TT_  4096
#define KK_  1024
#define BT_  65536
#define BETA_COMMIT 0.25f

#define ZQ_OFF     0ul
#define COMMIT_OFF 4194304ul
#define IDX_OFF    4194305ul
#define PERP_OFF   4259841ul
#define HIST_OFF   4259842ul

typedef __attribute__((ext_vector_type(2)))  float v2f;
typedef __attribute__((ext_vector_type(8)))  float v8f;

#if defined(__HIP_DEVICE_COMPILE__) && __has_builtin(__builtin_amdgcn_wmma_f32_16x16x4_f32)
#define WMMA_PATH 1   // fp32 WMMA (codegen-confirmed round 1): exact inner products
#else
#define WMMA_PATH 0
#endif

__device__ __forceinline__ v8f wmma4(v2f a, v2f b, v8f c) {
#if WMMA_PATH == 1
  return __builtin_amdgcn_wmma_f32_16x16x4_f32(
      /*neg_a=*/false, a, /*neg_b=*/false, b,
      /*c_mod=*/(short)0, c, /*reuse_a=*/false, /*reuse_b=*/false);
#else
  (void)a; (void)b;
  return c;   // host pass stub (never executed on device)
#endif
}

// ---------------------------------------------------------------------------
// Kernel 1: precompute e_sq[n] = ||E[n]||^2, zero histogram counters.
// ---------------------------------------------------------------------------
__global__ void __launch_bounds__(256)
vq_init(const float* __restrict__ emb, float* __restrict__ esq,
        unsigned* __restrict__ cnt) {
  int n = blockIdx.x * blockDim.x + threadIdx.x;
  if (n < KK_) {
    const float4* row = (const float4*)(emb + n * DD);
    float s = 0.f;
#pragma unroll
    for (int i = 0; i < DD / 4; ++i) {
      float4 v = row[i];
      s += v.x * v.x + v.y * v.y + v.z * v.z + v.w * v.w;
    }
    esq[n] = s;
    cnt[n] = 0u;
  }
}

// ---------------------------------------------------------------------------
// Kernel 2: main VQ kernel. 256 threads = 8 waves; each wave owns 16 rows
// (16 consecutive t within one batch). Block covers 128 rows -> 512 blocks.
// LDS: double-buffered 4-tile (64 codes x 64 dims = 16KB) chunks of E.
// ---------------------------------------------------------------------------
__global__ void __launch_bounds__(256)
vq_main(const float* __restrict__ z_e, const float* __restrict__ emb,
        const float* __restrict__ esq, unsigned* __restrict__ cnt,
        float* __restrict__ partials, float* __restrict__ out) {
  __shared__ float Btile[2][64 * DD];   // 2 x 16KB chunks (4 sub-tiles each)
  __shared__ int   idxbuf[8][16];

  const int tid  = threadIdx.x;
  const int wave = tid >> 5;
  const int lane = tid & 31;
  const int half = lane >> 4;        // 0: lanes 0-15, 1: lanes 16-31
  const int lrow = lane & 15;
  const int m0   = blockIdx.x * 128 + wave * 16;   // first of this wave's 16 rows
  const int b    = m0 >> 12;                       // m = b*T + t
  const int t    = (m0 & (TT_ - 1)) + lrow;        // this lane's t (rows coalesce)
  const float* zb = z_e + (size_t)b * (DD * TT_);

  // ---- preload A fragments: 16 rows x 64 dims, ISA 16x4-f32 striping ----
  // lane holds M = lrow; VGPR pair holds K = 4*kk + 2*half + {0,1}
  v2f afrag[16];
#pragma unroll
  for (int kk = 0; kk < 16; ++kk) {
    int d0 = 4 * kk + 2 * half;
    afrag[kk][0] = zb[d0 * TT_ + t];
    afrag[kk][1] = zb[(d0 + 1) * TT_ + t];
  }

  float best[8];
  int   bestn[8];
#pragma unroll
  for (int v = 0; v < 8; ++v) { best[v] = 3.4e38f; bestn[v] = 0; }

  // ---- sweep codebook in 16 chunks of 64 codes (4 x 16x16 sub-tiles) ----
  const float4* embv = (const float4*)emb;   // 16384 float4 total
  float4 pre[4];
#pragma unroll
  for (int i = 0; i < 4; ++i) pre[i] = embv[i * 256 + tid];   // chunk 0

  for (int it = 0; it < 16; ++it) {
    // store prefetched chunk into LDS buffer (opposite buffer from compute)
    float4* dst = (float4*)Btile[it & 1];
#pragma unroll
    for (int i = 0; i < 4; ++i) dst[i * 256 + tid] = pre[i];
    // issue next chunk's global loads now; latency hides under 64 WMMAs below
    if (it < 15) {
#pragma unroll
      for (int i = 0; i < 4; ++i) pre[i] = embv[(it + 1) * 1024 + i * 256 + tid];
    }
    __syncthreads();   // single barrier/iter: separates stores from same-buffer reads

    const float* bbase = &Btile[it & 1][lrow * DD];
#pragma unroll
    for (int st = 0; st < 4; ++st) {
      const int n = it * 64 + st * 16 + lrow;        // this lane's column
      const v2f* brow2 = (const v2f*)(bbase + st * (16 * DD));
      // 4 independent accumulators break the WMMA D->C RAW chain
      v8f a0 = {0.f,0.f,0.f,0.f,0.f,0.f,0.f,0.f};
      v8f a1 = a0, a2 = a0, a3 = a0;
#pragma unroll
      for (int kk = 0; kk < 16; kk += 4) {
        a0 = wmma4(afrag[kk],     brow2[2 * kk + half],           a0);
        a1 = wmma4(afrag[kk + 1], brow2[2 * (kk + 1) + half],     a1);
        a2 = wmma4(afrag[kk + 2], brow2[2 * (kk + 2) + half],     a2);
        a3 = wmma4(afrag[kk + 3], brow2[2 * (kk + 3) + half],     a3);
      }
      const float eq = esq[n];
#pragma unroll
      for (int v = 0; v < 8; ++v) {
        float s  = (a0[v] + a1[v]) + (a2[v] + a3[v]);
        float sc = fmaf(-2.f, s, eq);   // == dist - x_sq ; same argmin
        if (sc < best[v]) { best[v] = sc; bestn[v] = n; }  // n grows -> first hit kept
      }
    }
  }

  // ---- argmin across the 16 lanes of each half (rows 0-7 / 8-15) ----
#pragma unroll
  for (int off = 1; off <= 8; off <<= 1) {
#pragma unroll
    for (int v = 0; v < 8; ++v) {
      float ov = __shfl_xor(best[v], off, 32);
      int   oi = __shfl_xor(bestn[v], off, 32);
      if (ov < best[v] || (ov == best[v] && oi < bestn[v])) {
        best[v] = ov; bestn[v] = oi;   // tie-break: lowest index (match argmin)
      }
    }
  }
  if (lrow == 0) {
#pragma unroll
    for (int v = 0; v < 8; ++v) idxbuf[wave][half * 8 + v] = bestn[v];
  }
  __syncthreads();

  // ---- epilogue: indices, z_q_st scatter, commit partials, histogram ----
  const int idx = idxbuf[wave][lrow];        // all 32 lanes: row lrow's winner
  if (half == 0) {
    out[IDX_OFF + (size_t)(m0 + lrow)] = (float)idx;
    atomicAdd(&cnt[idx], 1u);
  }
  const float* eqrow = emb + (size_t)idx * DD;
  const size_t base  = (size_t)b * (DD * TT_) + (size_t)t;
  float lsum = 0.f;
#pragma unroll
  for (int dd = 0; dd < 32; ++dd) {
    int d = dd + 32 * half;                  // lanes 0-15 -> d 0..31, 16-31 -> 32..63
    float q = eqrow[d];
    size_t off = base + (size_t)d * TT_;
    float ze = __builtin_nontemporal_load(z_e + off);      // one-shot stream
    __builtin_nontemporal_store(q, out + ZQ_OFF + off);    // don't pollute caches
    float df = ze - q;
    lsum += df * df;
  }
#pragma unroll
  for (int off = 16; off > 0; off >>= 1) lsum += __shfl_xor(lsum, off, 32);
  if (lane == 0) partials[blockIdx.x * 8 + wave] = lsum;  // deterministic slot
}

// ---------------------------------------------------------------------------
// Kernel 3: usage_hist, perplexity, commit loss (deterministic reductions).
// ---------------------------------------------------------------------------
__global__ void __launch_bounds__(1024)
vq_final(const unsigned* __restrict__ cnt, const float* __restrict__ partials,
         float* __restrict__ out) {
  __shared__ float sh[1024];
  const int tid = threadIdx.x;

  // usage + entropy (total count is exactly BT_ = 65536 by construction)
  float p = (float)cnt[tid] * (1.0f / (float)BT_);
  out[HIST_OFF + tid] = p;
  sh[tid] = (p > 0.f) ? p * __logf(p) : 0.f;
  __syncthreads();
  for (int s = 512; s > 0; s >>= 1) {
    if (tid < s) sh[tid] += sh[tid + s];
    __syncthreads();
  }
  if (tid == 0) out[PERP_OFF] = __expf(-sh[0]);

  // commit loss: fixed-order sum of 512*8 = 4096 wave partials
  float ps = 0.f;
#pragma unroll
  for (int i = 0; i < 4; ++i) ps += partials[tid * 4 + i];
  __syncthreads();
  sh[tid] = ps;
  __syncthreads();
  for (int s = 512; s > 0; s >>= 1) {
    if (tid < s) sh[tid] += sh[tid + s];
    __syncthreads();
  }
  if (tid == 0)
    out[COMMIT_OFF] = BETA_COMMIT * sh[0] * (1.0f / (float)(BB * DD * TT_));
}

// ---------------------------------------------------------------------------
extern "C" void kernel_launch(void* const* d_in, const int* in_sizes, int n_in,
                              void* d_out, int out_size, void* d_ws, size_t ws_size,
                              hipStream_t stream) {
  const float* z_e = (const float*)d_in[0];   // (B, D, T) fp32
  const float* emb = (const float*)d_in[1];   // (K, D)   fp32
  float* out = (float*)d_out;

  float*    esq      = (float*)d_ws;                        // 1024 f32
  unsigned* cnt      = (unsigned*)((char*)d_ws + 4096);     // 1024 u32
  float*    partials = (float*)((char*)d_ws + 8192);        // 4096 f32

  vq_init <<<4,   256, 0, stream>>>(emb, esq, cnt);
  vq_main <<<512, 256, 0, stream>>>(z_e, emb, esq, cnt, partials, out);
  vq_final<<<1,  1024, 0, stream>>>(cnt, partials, out);
}